// HyperGNN_61100204753259
// MI455X (gfx1250) — compile-verified
//
#include <hip/hip_runtime.h>

typedef float v2f __attribute__((ext_vector_type(2)));
typedef float v4f __attribute__((ext_vector_type(4)));
typedef float v8f __attribute__((ext_vector_type(8)));

#define DIM     64
#define NA_     100000
#define NB_     100000
#define NEDGE_  1200000
#define WAVES   4

// D = A(16x4,f32) * B(4x16,f32) + C(16x16,f32) -- exact fp32 matrix op on CDNA5
static __device__ __forceinline__ v8f wmma4(v2f a, v2f b, v8f c) {
  return __builtin_amdgcn_wmma_f32_16x16x4_f32(
      /*neg_a=*/false, a, /*neg_b=*/false, b,
      /*c_mod=*/(short)0, c, /*reuse_a=*/false, /*reuse_b=*/false);
}

__global__ void zero_f32(float* __restrict__ p, int n) {
  int i = blockIdx.x * blockDim.x + threadIdx.x;
  if (i < n) p[i] = 0.0f;
}

// One thread per (edge, feature). Sources & accumulators are L2-resident
// (25.6 MB each vs 192 MB L2), so atomics hit L2 at full rate.
__global__ void scatter_accum(const float* __restrict__ x,
                              const int* __restrict__ src,
                              const int* __restrict__ dst,
                              float* __restrict__ s,
                              float* __restrict__ cnt) {
  long idx = (long)blockIdx.x * blockDim.x + threadIdx.x;
  int e = (int)(idx >> 6);
  if (e >= NEDGE_) return;
  int f = (int)(idx & 63);
  int sv = src[e];
  int dv = dst[e];
  float val = x[sv * DIM + f];
  atomicAdd(s + dv * DIM + f, val);
  if (f == 0) atomicAdd(cnt + dv, 1.0f);
}

// new_B = (S0/max(C0,1)) @ Wn + xB @ Wr + bias ; one wave per 16-node tile
__global__ __launch_bounds__(32 * WAVES)
void update_B_kernel(const float* __restrict__ xB,
                     const float* __restrict__ s0, const float* __restrict__ c0,
                     const float* __restrict__ Wn, const float* __restrict__ Wr,
                     const float* __restrict__ bias,
                     float* __restrict__ outB) {
  __shared__ float lds[WAVES * 2 * 16 * DIM];
  int wave = threadIdx.x >> 5;
  int lane = threadIdx.x & 31;
  int tile = blockIdx.x * WAVES + wave;
  bool active = (tile < NB_ / 16);
  float* ldsM = lds + wave * 2 * 16 * DIM;
  float* ldsX = ldsM + 16 * DIM;
  if (active) {
    int base = tile * 16;
    for (int i = lane; i < 16 * 16; i += 32) {
      int row = i >> 4;
      int c4  = (i & 15) * 4;
      v4f xv = *(const v4f*)(xB + (base + row) * DIM + c4);
      *(v4f*)(ldsX + row * DIM + c4) = xv;
      float inv = 1.0f / fmaxf(c0[base + row], 1.0f);
      v4f sv = *(const v4f*)(s0 + (base + row) * DIM + c4);
      sv.x *= inv; sv.y *= inv; sv.z *= inv; sv.w *= inv;
      *(v4f*)(ldsM + row * DIM + c4) = sv;
    }
  }
  __syncthreads();
  if (!active) return;
  int base = tile * 16;
  int half = lane >> 4;      // selects K pair / D-row half per ISA layout
  int n    = lane & 15;      // A: M-row ; B/D: N-col
  for (int nc = 0; nc < DIM; nc += 16) {
    float bv = bias[nc + n];
    v8f acc;
#pragma unroll
    for (int v = 0; v < 8; ++v) acc[v] = bv;
#pragma unroll
    for (int k0 = 0; k0 < DIM; k0 += 4) {
      int kk = k0 + 2 * half;
      v2f aM = *(const v2f*)(ldsM + n * DIM + kk);
      v2f aX = *(const v2f*)(ldsX + n * DIM + kk);
      v2f bN, bR;
      bN.x = Wn[kk * DIM + nc + n];
      bN.y = Wn[(kk + 1) * DIM + nc + n];
      bR.x = Wr[kk * DIM + nc + n];
      bR.y = Wr[(kk + 1) * DIM + nc + n];
      acc = wmma4(aM, bN, acc);
      acc = wmma4(aX, bR, acc);
    }
#pragma unroll
    for (int v = 0; v < 8; ++v)
      outB[(base + v + 8 * half) * DIM + nc + n] = acc[v];
  }
}

// new_A = M1@Wn1 + M2@Wn2 + xA@(Wr1+Wr2) + (b1+b2)
__global__ __launch_bounds__(32 * WAVES)
void update_A_kernel(const float* __restrict__ xA,
                     const float* __restrict__ s1, const float* __restrict__ c1,
                     const float* __restrict__ s2, const float* __restrict__ c2,
                     const float* __restrict__ Wn1, const float* __restrict__ Wr1,
                     const float* __restrict__ Wn2, const float* __restrict__ Wr2,
                     const float* __restrict__ b1,  const float* __restrict__ b2,
                     float* __restrict__ outA) {
  __shared__ float lds[WAVES * 3 * 16 * DIM];
  int wave = threadIdx.x >> 5;
  int lane = threadIdx.x & 31;
  int tile = blockIdx.x * WAVES + wave;
  bool active = (tile < NA_ / 16);
  float* ldsM1 = lds + wave * 3 * 16 * DIM;
  float* ldsX  = ldsM1 + 16 * DIM;
  float* ldsM2 = ldsX  + 16 * DIM;
  if (active) {
    int base = tile * 16;
    for (int i = lane; i < 16 * 16; i += 32) {
      int row = i >> 4;
      int c4  = (i & 15) * 4;
      v4f xv = *(const v4f*)(xA + (base + row) * DIM + c4);
      *(v4f*)(ldsX + row * DIM + c4) = xv;
      float inv1 = 1.0f / fmaxf(c1[base + row], 1.0f);
      v4f m1 = *(const v4f*)(s1 + (base + row) * DIM + c4);
      m1.x *= inv1; m1.y *= inv1; m1.z *= inv1; m1.w *= inv1;
      *(v4f*)(ldsM1 + row * DIM + c4) = m1;
      float inv2 = 1.0f / fmaxf(c2[base + row], 1.0f);
      v4f m2 = *(const v4f*)(s2 + (base + row) * DIM + c4);
      m2.x *= inv2; m2.y *= inv2; m2.z *= inv2; m2.w *= inv2;
      *(v4f*)(ldsM2 + row * DIM + c4) = m2;
    }
  }
  __syncthreads();
  if (!active) return;
  int base = tile * 16;
  int half = lane >> 4;
  int n    = lane & 15;
  for (int nc = 0; nc < DIM; nc += 16) {
    float bv = b1[nc + n] + b2[nc + n];
    v8f acc;
#pragma unroll
    for (int v = 0; v < 8; ++v) acc[v] = bv;
#pragma unroll
    for (int k0 = 0; k0 < DIM; k0 += 4) {
      int kk = k0 + 2 * half;
      v2f a1 = *(const v2f*)(ldsM1 + n * DIM + kk);
      v2f aX = *(const v2f*)(ldsX  + n * DIM + kk);
      v2f a2 = *(const v2f*)(ldsM2 + n * DIM + kk);
      v2f bn1, brS, bn2;
      bn1.x = Wn1[kk * DIM + nc + n];
      bn1.y = Wn1[(kk + 1) * DIM + nc + n];
      brS.x = Wr1[kk * DIM + nc + n] + Wr2[kk * DIM + nc + n];
      brS.y = Wr1[(kk + 1) * DIM + nc + n] + Wr2[(kk + 1) * DIM + nc + n];
      bn2.x = Wn2[kk * DIM + nc + n];
      bn2.y = Wn2[(kk + 1) * DIM + nc + n];
      acc = wmma4(a1, bn1, acc);
      acc = wmma4(aX, brS, acc);
      acc = wmma4(a2, bn2, acc);
    }
#pragma unroll
    for (int v = 0; v < 8; ++v)
      outA[(base + v + 8 * half) * DIM + nc + n] = acc[v];
  }
}

// out = relu(x @ W_out + b_out), A rows then B rows
__global__ __launch_bounds__(32 * WAVES)
void head_kernel(const float* __restrict__ xA, const float* __restrict__ xB,
                 const float* __restrict__ W,  const float* __restrict__ bo,
                 float* __restrict__ out) {
  int wave = threadIdx.x >> 5;
  int lane = threadIdx.x & 31;
  int tile = blockIdx.x * WAVES + wave;
  const int ntiles = (NA_ + NB_) / 16;
  if (tile >= ntiles) return;
  const float* x;
  int base;
  if (tile < NA_ / 16) { x = xA; base = tile * 16; }
  else                 { x = xB; base = (tile - NA_ / 16) * 16; }
  int obase = tile * 16;
  int half = lane >> 4;
  int n    = lane & 15;
  for (int nc = 0; nc < DIM; nc += 16) {
    float bv = bo[nc + n];
    v8f acc;
#pragma unroll
    for (int v = 0; v < 8; ++v) acc[v] = bv;
#pragma unroll
    for (int k0 = 0; k0 < DIM; k0 += 4) {
      int kk = k0 + 2 * half;
      v2f a = *(const v2f*)(x + (base + n) * DIM + kk);
      v2f b;
      b.x = W[kk * DIM + nc + n];
      b.y = W[(kk + 1) * DIM + nc + n];
      acc = wmma4(a, b, acc);
    }
#pragma unroll
    for (int v = 0; v < 8; ++v)
      out[(obase + v + 8 * half) * DIM + nc + n] = fmaxf(acc[v], 0.0f);
  }
}

extern "C" void kernel_launch(void* const* d_in, const int* in_sizes, int n_in,
                              void* d_out, int out_size, void* d_ws, size_t ws_size,
                              hipStream_t stream) {
  const float* xA   = (const float*)d_in[0];
  const float* xB   = (const float*)d_in[1];
  const float* Wn   = (const float*)d_in[2];   // [2,3,64,64]
  const float* Wr   = (const float*)d_in[3];   // [2,3,64,64]
  const float* b    = (const float*)d_in[4];   // [2,3,64]
  const float* Wout = (const float*)d_in[5];   // [64,64]
  const float* bout = (const float*)d_in[6];   // [64]
  const int* src0 = (const int*)d_in[7];
  const int* dst0 = (const int*)d_in[8];
  const int* src1 = (const int*)d_in[9];
  const int* dst1 = (const int*)d_in[10];
  const int* src2 = (const int*)d_in[11];
  const int* dst2 = (const int*)d_in[12];
  float* out = (float*)d_out;

  // workspace layout (floats): ping-pong activations + per-layer accumulators
  float* ws = (float*)d_ws;
  size_t off = 0;
  float* A0 = ws + off; off += (size_t)NA_ * DIM;
  float* B0 = ws + off; off += (size_t)NB_ * DIM;
  float* A1 = ws + off; off += (size_t)NA_ * DIM;
  float* B1 = ws + off; off += (size_t)NB_ * DIM;
  float* S0 = ws + off; off += (size_t)NB_ * DIM;   // S0..C2 contiguous for one zero pass
  float* S1 = ws + off; off += (size_t)NA_ * DIM;
  float* S2 = ws + off; off += (size_t)NA_ * DIM;
  float* C0 = ws + off; off += NB_;
  float* C1 = ws + off; off += NA_;
  float* C2 = ws + off; off += NA_;

  const int zn = NB_ * DIM + 2 * (NA_ * DIM) + NB_ + 2 * NA_;  // S0..C2 span
  dim3 blk256(256), blkW(32 * WAVES);
  int scatterBlocks = (int)(((long)NEDGE_ * DIM + 255) / 256);
  int updBlocks  = (NA_ / 16 + WAVES - 1) / WAVES;              // NA_ == NB_
  int headBlocks = ((NA_ + NB_) / 16 + WAVES - 1) / WAVES;

  const float* inA = xA;
  const float* inB = xB;
  float* outAbuf[2] = {A0, A1};
  float* outBbuf[2] = {B0, B1};

  for (int l = 0; l < 2; ++l) {
    zero_f32<<<(zn + 255) / 256, blk256, 0, stream>>>(S0, zn);
    scatter_accum<<<scatterBlocks, blk256, 0, stream>>>(inA, src0, dst0, S0, C0);
    scatter_accum<<<scatterBlocks, blk256, 0, stream>>>(inB, src1, dst1, S1, C1);
    scatter_accum<<<scatterBlocks, blk256, 0, stream>>>(inA, src2, dst2, S2, C2);

    update_B_kernel<<<updBlocks, blkW, 0, stream>>>(
        inB, S0, C0,
        Wn + (size_t)(l * 3 + 0) * DIM * DIM,
        Wr + (size_t)(l * 3 + 0) * DIM * DIM,
        b  + (size_t)(l * 3 + 0) * DIM,
        outBbuf[l]);
    update_A_kernel<<<updBlocks, blkW, 0, stream>>>(
        inA, S1, C1, S2, C2,
        Wn + (size_t)(l * 3 + 1) * DIM * DIM,
        Wr + (size_t)(l * 3 + 1) * DIM * DIM,
        Wn + (size_t)(l * 3 + 2) * DIM * DIM,
        Wr + (size_t)(l * 3 + 2) * DIM * DIM,
        b  + (size_t)(l * 3 + 1) * DIM,
        b  + (size_t)(l * 3 + 2) * DIM,
        outAbuf[l]);
    inA = outAbuf[l];
    inB = outBbuf[l];
  }

  head_kernel<<<headBlocks, blkW, 0, stream>>>(inA, inB, Wout, bout, out);
}